// Retinal_NET_50697793962440
// MI455X (gfx1250) — compile-verified
//
#include <hip/hip_runtime.h>
#include <hip/hip_bf16.h>

typedef __attribute__((ext_vector_type(2))) float v2f;
typedef __attribute__((ext_vector_type(8))) float v8f;
typedef unsigned int u32;
typedef __attribute__((ext_vector_type(4))) u32 v4u;
typedef __attribute__((ext_vector_type(8))) int v8i;
typedef __attribute__((ext_vector_type(4))) int v4i;

// Workgroup per (batch b, flat cell p). Pipeline:
//   TDM DMA  : x tile (29 t x 25 rows x 25 cols) -> LDS (tensor_load_to_lds)
//   s1[tau][pos]  : stage-1 spatial conv from LDS, tau = t-31 in [0,29)
//   first[tt][pos]: causal 15-tap temporal conv via V_WMMA_F32_16X16X4_F32 + ReLU
//   flat[b][tt][p]: 25-tap last conv reduce
__global__ __launch_bounds__(256) void retina_stage1(
    const float* __restrict__ x,          // (16,1,60,250,250)
    const float* __restrict__ space_w,    // 25
    const float* __restrict__ temporal_w, // 15
    const float* __restrict__ last_w,     // 25
    float* __restrict__ flat_ws)          // (16,15,100)
{
    __shared__ float xt[29][25][25];  // 72.5 KB input tile (TDM destination)
    __shared__ float s1[32][32];      // [tau][pos], zero padded to WMMA tile size
    __shared__ float fi[16][32];      // relu(first) [tt][pos]
    __shared__ float twl[64];         // zero-padded band: twl[16+k] = tw[k]

    const int tid = threadIdx.x;
    const int g   = blockIdx.x;
    const int b   = g / 100;
    const int p   = g % 100;
    const int H0  = (p / 10) * 25;
    const int W0  = (p % 10) * 25;
    const int wid = tid >> 5;

    const float* tile_base =
        x + (((long)b * 60 + 31) * 250 + H0) * 250 + W0;

#if __has_builtin(__builtin_amdgcn_tensor_load_to_lds)
    // ---- Tensor Data Mover: DMA the 25x25x29 f32 tile into LDS ----
    if (wid == 0) {   // wave-uniform: exactly one TDM issue per workgroup
        const u32 lds_off = (u32)(size_t)(&xt[0][0][0]);
        const unsigned long long ga = (unsigned long long)(size_t)tile_base;

        v4u g0;
        g0.x = 1u;                                   // count=1, user descriptor
        g0.y = lds_off;                              // LDS byte address of tile
        g0.z = (u32)ga;                              // global_addr[31:0]
        g0.w = ((u32)(ga >> 32) & 0x01FFFFFFu)       // global_addr[56:32]
               | (2u << 30);                         // type = 2 ("image")

        v8i g1;
        g1[0] = 0x00020000;          // wg_mask=0, data_size=2 (4B), no pad/iter
        g1[1] = 0x00FA0000;          // tensor_dim0 = 250 (bits 79:48)
        g1[2] = 0x00FA0000;          // tensor_dim1 = 250 (bits 111:80)
        g1[3] = 0x00190000;          // tile_dim0 = 25 (bits 127:112)
        g1[4] = 0x001D0019;          // tile_dim1 = 25 | tile_dim2 = 29 << 16
        g1[5] = 250;                 // tensor_dim0_stride = 250 (row stride)
        g1[6] = (int)0xF4240000;     // tensor_dim1_stride[15:0]=62500 (slice stride)
        g1[7] = 0;                   // tensor_dim1_stride[47:16] = 0

        v4i g2;                      // tensor_dim2=29; dim3 unused (3D tile)
        g2[0] = 29; g2[1] = 0; g2[2] = 0; g2[3] = 0;
        v4i g3;
        g3[0] = 0; g3[1] = 0; g3[2] = 0; g3[3] = 0;

#if defined(__clang_major__) && (__clang_major__ >= 23)
        v8i g4;
        g4[0]=0; g4[1]=0; g4[2]=0; g4[3]=0; g4[4]=0; g4[5]=0; g4[6]=0; g4[7]=0;
        __builtin_amdgcn_tensor_load_to_lds(g0, g1, g2, g3, g4, 0);
#else
        __builtin_amdgcn_tensor_load_to_lds(g0, g1, g2, g3, 0);
#endif
    }
    __builtin_amdgcn_s_wait_tensorcnt(0);   // TENSORcnt==0 trivially on waves 1..7
#else
    // ---- Fallback: cooperative global -> LDS copy ----
    for (int i = tid; i < 29 * 625; i += 256) {
        const int tau = i / 625;
        const int rem = i % 625;
        (&xt[0][0][0])[i] = tile_base[(long)tau * 62500 + (rem / 25) * 250 + (rem % 25)];
    }
#endif

    // zero the padded s1 tile (rows 29..31 / cols 25..31 must be 0 for WMMA)
    for (int i = tid; i < 32 * 32; i += 256) (&s1[0][0])[i] = 0.f;
    // zero-padded temporal band table (branch-free A-matrix gather)
    if (tid < 64) {
        const int k = tid - 16;
        twl[tid] = (k >= 0 && k < 15) ? temporal_w[k] : 0.f;
    }
    __syncthreads();

    float sw[25];
#pragma unroll
    for (int i = 0; i < 25; ++i) sw[i] = space_w[i];

    // Stage 1: 29 timesteps x 25 positions, disjoint 5x5 patches read from LDS
    for (int idx = tid; idx < 29 * 25; idx += 256) {
        const int tau = idx / 25;
        const int pos = idx % 25;
        const int h5  = pos / 5, w5 = pos % 5;
        float acc = 0.f;
#pragma unroll
        for (int r = 0; r < 5; ++r)
#pragma unroll
            for (int c = 0; c < 5; ++c)
                acc += sw[r * 5 + c] * xt[tau][5 * h5 + r][5 * w5 + c];
        s1[tau][pos] = acc;
    }
    __syncthreads();

    // Stage 2: first = TW(16x32 banded) x s1(32x32-padded), V_WMMA_F32_16X16X4_F32.
    // Waves 0 and 1 each own one 16-wide N tile; branch is wave-uniform (EXEC all-1s).
    if (wid < 2) {
        const int lane  = tid & 31;
        const int hi    = lane >> 4;   // half-wave select (K pairing per ISA layout)
        const int mn    = lane & 15;   // M for A, N for B/D
        const int nbase = wid * 16;
        const int abase = 16 + 2 * hi - mn;  // twl index of A[m, 2*hi]

        v8f c;
#pragma unroll
        for (int r = 0; r < 8; ++r) c[r] = 0.f;

#pragma unroll
        for (int kk = 0; kk < 8; ++kk) {
            const int t0 = 4 * kk + 2 * hi;
            v2f a, bb;
            a.x = twl[abase + 4 * kk];       // A[m,tau] = tw[tau-m] (banded)
            a.y = twl[abase + 4 * kk + 1];
            bb.x = s1[t0][nbase + mn];
            bb.y = s1[t0 + 1][nbase + mn];
            c = __builtin_amdgcn_wmma_f32_16x16x4_f32(
                    false, a, false, bb, (short)0, c, false, false);
        }
#pragma unroll
        for (int r = 0; r < 8; ++r) {
            float v = c[r];
            fi[r + 8 * hi][nbase + mn] = v > 0.f ? v : 0.f;   // ReLU
        }
    }
    __syncthreads();

    // Stage 3: last conv (25-tap reduce over pos) -> flat[b][tt][p], tt = 0..14
    if (tid < 15) {
        float acc = 0.f;
#pragma unroll
        for (int pos = 0; pos < 25; ++pos) acc += last_w[pos] * fi[tid][pos];
        flat_ws[((long)b * 15 + tid) * 100 + p] = acc;
    }
}

__device__ __forceinline__ float sigmoidf(float v) {
    return 1.f / (1.f + __expf(-v));
}

// One block per batch: ama/gang temporal convs at t=59 only, then 49-tap column reduce.
__global__ __launch_bounds__(64) void retina_final(
    const float* __restrict__ flat_ws,    // (16,15,100)
    const float* __restrict__ acw_, const float* __restrict__ akw,
    const float* __restrict__ ab_,  const float* __restrict__ alpha_,
    const float* __restrict__ gcw_, const float* __restrict__ gkw,
    const float* __restrict__ gb_,  const float* __restrict__ colw,
    float* __restrict__ out)
{
    __shared__ float red[49];
    const int b = blockIdx.x;
    const int j = threadIdx.x;
    if (j < 49) {
        const float acw = acw_[0], ab = ab_[0], alpha = alpha_[0];
        const float gcw = gcw_[0], gb = gb_[0];
        const int ia = 2 * j;                 // ama taps even flat positions
        const int ig = (2 * j + 25) % 100;    // gang taps rolled positions
        float ua = ab, ug = gb;
#pragma unroll
        for (int k = 0; k < 15; ++k) {
            const float* f = flat_ws + ((long)b * 15 + k) * 100;
            ua += akw[k] * (acw * f[ia]);
            ug += gkw[k] * (gcw * f[ig]);
        }
        const float ama = alpha * sigmoidf(ua);
        const float gt  = sigmoidf(ug - fabsf(ama));
        red[j] = colw[j] * gt;
    }
    __syncthreads();
    if (j == 0) {
        float s = 0.f;
#pragma unroll
        for (int k = 0; k < 49; ++k) s += red[k];
        out[b] = s;
    }
}

extern "C" void kernel_launch(void* const* d_in, const int* in_sizes, int n_in,
                              void* d_out, int out_size, void* d_ws, size_t ws_size,
                              hipStream_t stream) {
    const float* x          = (const float*)d_in[0];
    const float* space_w    = (const float*)d_in[1];
    const float* temporal_w = (const float*)d_in[2];
    const float* last_w     = (const float*)d_in[3];
    const float* ama_create = (const float*)d_in[4];
    const float* ama_kernel = (const float*)d_in[5];
    const float* ama_bias   = (const float*)d_in[6];
    const float* ama_alpha  = (const float*)d_in[7];
    const float* gang_create= (const float*)d_in[8];
    const float* gang_kernel= (const float*)d_in[9];
    const float* gang_bias  = (const float*)d_in[10];
    const float* gang_col   = (const float*)d_in[11];
    float* out = (float*)d_out;
    float* flat_ws = (float*)d_ws;   // 16*15*100 floats = 96 KB

    retina_stage1<<<1600, 256, 0, stream>>>(x, space_w, temporal_w, last_w, flat_ws);
    retina_final<<<16, 64, 0, stream>>>(flat_ws, ama_create, ama_kernel, ama_bias,
                                        ama_alpha, gang_create, gang_kernel, gang_bias,
                                        gang_col, out);
}